// AucLoss_3289944948924
// MI455X (gfx1250) — compile-verified
//
#include <hip/hip_runtime.h>

// AUC pairwise-margin loss, MI455X (gfx1250, wave32).
// Phase 1: streaming row-max of x (67 MB, HBM-bound ~2.9us) with NT b128 loads.
// Phase 2: 8192x8192 conditional square-sum, neg slice staged in LDS via
//          global_load_async_to_lds_b128 + s_wait_asynccnt (gfx1250 async path).
// Phase 3: deterministic final reduction (no FP atomics anywhere).

typedef float v4f __attribute__((ext_vector_type(4)));

#define BN   8192      // batch
#define CN   2048      // classes
#define GAMA 0.3f
#define INVALID_P 3.0e30f
#define NJ   8         // j-slices in phase 2
#define JT   (BN / NJ) // 1024 neg values per slice

__device__ __forceinline__ float sigmoidf_fast(float v) {
    return 1.0f / (1.0f + __expf(-v));
}

// ---------------- Phase 1: per-row masked max + pos gather ----------------
__global__ __launch_bounds__(256)
void row_reduce_kernel(const float* __restrict__ x, const int* __restrict__ y,
                       float* __restrict__ p_out, float* __restrict__ n_out) {
    const int lane = threadIdx.x & 31;
    const int wave = threadIdx.x >> 5;
    const int row  = blockIdx.x * 8 + wave;           // one row per wave32
    const float* xr = x + (size_t)row * CN;
    const int lbl = y[row];

    float m = -3.4e38f;
#pragma unroll
    for (int k = 0; k < CN / 128; ++k) {              // 16 x b128 NT loads/lane
        const int c0 = k * 128 + lane * 4;
        v4f v = __builtin_nontemporal_load((const v4f*)(xr + c0));
#pragma unroll
        for (int j = 0; j < 4; ++j) {
            const int c = c0 + j;
            const bool excl = (c == lbl) || (c == 0); // label col and unk col
            m = excl ? m : fmaxf(m, v[j]);
        }
    }
    // wave32 max reduction
#pragma unroll
    for (int off = 16; off > 0; off >>= 1)
        m = fmaxf(m, __shfl_xor(m, off, 32));

    if (lane == 0) {
        const bool valid = (lbl != 0);
        const float sp = sigmoidf_fast(xr[lbl]);      // only 2 exps per row
        const float sn = sigmoidf_fast(m);            // sigma(max) == max(sigma)
        p_out[row] = valid ? (sp - GAMA) : INVALID_P; // invalid -> never kept
        n_out[row] = sn;
    }
}

// ---------------- Phase 2: pairwise conditional square-sum ----------------
__global__ __launch_bounds__(256)
void pairwise_kernel(const float* __restrict__ p_arr,
                     const float* __restrict__ n_arr,
                     float* __restrict__ partials) {
    __shared__ float nsh[JT];                         // 4 KB slice of neg
    __shared__ float red[8];

    const int tid = threadIdx.x;
    const int i   = blockIdx.x * 256 + tid;           // this thread's pos row
    const int j0  = blockIdx.y * JT;                  // neg slice base

    // gfx1250 async copy: each lane DMAs 16B global -> LDS, no VGPR staging.
    {
        const int k = tid * 4;                        // 256 lanes x 4 floats = JT
        const float* gsrc = n_arr + j0 + k;
        const unsigned ldst = (unsigned)(size_t)(&nsh[k]);
        asm volatile("global_load_async_to_lds_b128 %0, %1, off"
                     :: "v"(ldst), "v"(gsrc) : "memory");
    }
    asm volatile("s_wait_asynccnt 0" ::: "memory");
    __syncthreads();

    const float p = p_arr[i];
    float acc = 0.0f;
#pragma unroll 8
    for (int j = 0; j < JT; ++j) {
        const float mg = p - nsh[j];                  // margin (gamma folded in p)
        const float t  = fminf(mg, 0.0f);             // keep only margin < 0
        acc = fmaf(t, t, acc);
    }

    // deterministic block reduction: wave32 shuffle tree + LDS across 8 waves
#pragma unroll
    for (int off = 16; off > 0; off >>= 1) acc += __shfl_xor(acc, off, 32);
    const int lane = tid & 31, w = tid >> 5;
    if (lane == 0) red[w] = acc;
    __syncthreads();
    if (tid < 32) {
        float s = (lane < 8) ? red[lane] : 0.0f;
#pragma unroll
        for (int off = 4; off > 0; off >>= 1) s += __shfl_xor(s, off, 32);
        if (tid == 0) partials[blockIdx.y * gridDim.x + blockIdx.x] = s;
    }
}

// ---------------- Phase 3: finalize (sum partials, count valid) ----------------
__global__ __launch_bounds__(256)
void finalize_kernel(const float* __restrict__ p_arr,
                     const float* __restrict__ partials,
                     float* __restrict__ out) {
    __shared__ float redf[8];
    __shared__ int   redi[8];
    const int tid = threadIdx.x;

    float s = partials[tid];                          // exactly 256 partials
    int cnt = 0;
    for (int i = tid; i < BN; i += 256)
        cnt += (p_arr[i] < 1.0e29f) ? 1 : 0;          // valid rows

#pragma unroll
    for (int off = 16; off > 0; off >>= 1) {
        s   += __shfl_xor(s, off, 32);
        cnt += __shfl_xor(cnt, off, 32);
    }
    const int lane = tid & 31, w = tid >> 5;
    if (lane == 0) { redf[w] = s; redi[w] = cnt; }
    __syncthreads();
    if (tid < 32) {
        s   = (lane < 8) ? redf[lane] : 0.0f;
        cnt = (lane < 8) ? redi[lane] : 0;
#pragma unroll
        for (int off = 4; off > 0; off >>= 1) {
            s   += __shfl_xor(s, off, 32);
            cnt += __shfl_xor(cnt, off, 32);
        }
        if (tid == 0)
            out[0] = s / ((float)cnt + 1.0f) / ((float)BN + 1.0f);
    }
}

extern "C" void kernel_launch(void* const* d_in, const int* in_sizes, int n_in,
                              void* d_out, int out_size, void* d_ws, size_t ws_size,
                              hipStream_t stream) {
    const float* x = (const float*)d_in[0];
    const int*   y = (const int*)d_in[1];
    float* ws       = (float*)d_ws;
    float* p_arr    = ws;             // [BN]
    float* n_arr    = ws + BN;        // [BN]
    float* partials = ws + 2 * BN;    // [NJ * 32] = 256

    row_reduce_kernel<<<BN / 8, 256, 0, stream>>>(x, y, p_arr, n_arr);
    dim3 g2(BN / 256, NJ);            // 32 x 8 blocks
    pairwise_kernel<<<g2, 256, 0, stream>>>(p_arr, n_arr, partials);
    finalize_kernel<<<1, 256, 0, stream>>>(p_arr, partials, (float*)d_out);
}